// LIFLayer_16020228014950
// MI455X (gfx1250) — compile-verified
//
#include <hip/hip_runtime.h>
#include <stdint.h>

// LIF scan: [T, B, N] with T=512, B=32, N=4096 (from reference).
#define T_STEPS  512
#define BN_ELEMS (32 * 4096)        // floats per timestep = 131072
#define THREADS  256                // 8 waves / block (wave32)
#define PIPE     8                  // async global->LDS pipeline depth
#define PF_DIST  32                 // L2 prefetch distance (timesteps)

typedef float v4f __attribute__((ext_vector_type(4)));

// One LIF update, IEEE op-order identical to the JAX reference:
//   V = V + (I - V)/20 ; spike = V>=1 ; V = spike ? 0 : V
__device__ __forceinline__ void lif_step(float& v, float x, float& s_out, float& w_out) {
    v = v + (x - v) / 20.0f;            // exact IEEE divide (matches XLA)
    const bool sp = (v >= 1.0f);
    s_out = sp ? 1.0f : 0.0f;
    v     = sp ? 0.0f : v;
    w_out = v;                          // post-reset voltage trace
}

__global__ __launch_bounds__(THREADS)
void lif_scan_kernel(const float* __restrict__ in,
                     float* __restrict__ spk,
                     float* __restrict__ vol)
{
    __shared__ v4f buf[PIPE][THREADS];  // 8 x 4KB buffer ring (32 KB)

    const int    tid  = threadIdx.x;
    const int    col4 = blockIdx.x * THREADS + tid;   // float4 column
    const size_t base = (size_t)col4 * 4;             // float offset in [B*N)

    // LDS byte offset of this thread's slot 0. Generic->flat truncation is the
    // LDS offset (LDS aperture is 4GB aligned; addr[31:0] == ds offset).
    const uint32_t lds0 = (uint32_t)(uintptr_t)(&buf[0][0]) + (uint32_t)tid * 16u;

    float v0 = 0.0f, v1 = 0.0f, v2 = 0.0f, v3 = 0.0f;

    // ---- prologue: fill the ASYNCcnt pipeline (tiles t = 0..PIPE-1) ----
#pragma unroll
    for (int p = 0; p < PIPE; ++p) {
        const uint32_t lds = lds0 + (uint32_t)p * (THREADS * 16u);
        const uint64_t ga  = (uint64_t)(uintptr_t)(in + (size_t)p * BN_ELEMS + base);
        asm volatile("global_load_async_to_lds_b128 %0, %1, off"
                     :: "v"(lds), "v"(ga) : "memory");
    }

    const int TMAIN = T_STEPS - PIPE;

    for (int t = 0; t < TMAIN; ++t) {
        // Pull the stream into L2 well ahead of the async engine.
        if (t + PF_DIST < T_STEPS)
            __builtin_prefetch(in + (size_t)(t + PF_DIST) * BN_ELEMS + base, 0, 1);

        // Async loads retire in order: <=7 outstanding => tile t is in LDS.
        asm volatile("s_wait_asynccnt 7" ::: "memory");

        const int slot = t & (PIPE - 1);
        v4f x = buf[slot][tid];                       // ds_load_b128

        // Refill the slot for tile t+PIPE. Fence DScnt so the ds_load above
        // has drained LDS before the async engine may overwrite the slot.
        {
            const uint32_t lds = lds0 + (uint32_t)slot * (THREADS * 16u);
            const uint64_t ga  = (uint64_t)(uintptr_t)(in + (size_t)(t + PIPE) * BN_ELEMS + base);
            asm volatile("s_wait_dscnt 0\n\t"
                         "global_load_async_to_lds_b128 %0, %1, off"
                         :: "v"(lds), "v"(ga) : "memory");
        }

        float s0, s1, s2, s3, w0, w1, w2, w3;
        lif_step(v0, x.x, s0, w0);
        lif_step(v1, x.y, s1, w1);
        lif_step(v2, x.z, s2, w2);
        lif_step(v3, x.w, s3, w3);
        v4f s = (v4f){s0, s1, s2, s3};
        v4f w = (v4f){w0, w1, w2, w3};

        const size_t o = (size_t)t * BN_ELEMS + base;
        __builtin_nontemporal_store(s, (v4f*)(spk + o));   // write-once traces:
        __builtin_nontemporal_store(w, (v4f*)(vol + o));   // nt stores
    }

    // ---- epilogue: drain pipeline, consume last PIPE tiles ----
    asm volatile("s_wait_asynccnt 0" ::: "memory");
#pragma unroll
    for (int t = TMAIN; t < T_STEPS; ++t) {
        v4f x = buf[t & (PIPE - 1)][tid];
        float s0, s1, s2, s3, w0, w1, w2, w3;
        lif_step(v0, x.x, s0, w0);
        lif_step(v1, x.y, s1, w1);
        lif_step(v2, x.z, s2, w2);
        lif_step(v3, x.w, s3, w3);
        v4f s = (v4f){s0, s1, s2, s3};
        v4f w = (v4f){w0, w1, w2, w3};
        const size_t o = (size_t)t * BN_ELEMS + base;
        __builtin_nontemporal_store(s, (v4f*)(spk + o));
        __builtin_nontemporal_store(w, (v4f*)(vol + o));
    }
}

extern "C" void kernel_launch(void* const* d_in, const int* in_sizes, int n_in,
                              void* d_out, int out_size, void* d_ws, size_t ws_size,
                              hipStream_t stream) {
    (void)in_sizes; (void)n_in; (void)out_size; (void)d_ws; (void)ws_size;
    const float* in  = (const float*)d_in[0];
    float*       spk = (float*)d_out;                                // spike trace
    float*       vol = spk + (size_t)T_STEPS * BN_ELEMS;             // voltage trace

    dim3 grid(BN_ELEMS / (THREADS * 4));   // 128 blocks x 256 threads, 4 floats/thread
    dim3 block(THREADS);
    lif_scan_kernel<<<grid, block, 0, stream>>>(in, spk, vol);
}